// focus_17738214932758
// MI455X (gfx1250) — compile-verified
//
#include <hip/hip_runtime.h>
#include <math.h>

#define DIMS   1024
#define HEADS  16
#define HD     64
#define NB     2
#define CTX    1536
#define WN     24            // windows per batch (CTX/HD)
#define NWIN   (NB*WN)       // 48
#define ROWS   (NB*CTX)      // 3072
#define EPSF   1e-6f
#define NUMEL_INV (1.0f/3145728.0f)   // b*h*c*hd, same for stage1 & stage2

typedef float v2f __attribute__((ext_vector_type(2)));
typedef float v8f __attribute__((ext_vector_type(8)));

// ---------------------------------------------------------------- CDNA5 async copy
// GLOBAL_LOAD_ASYNC_TO_LDS_B128 (ISA 15.18.3 op 98): copies 16 bytes from global memory into LDS,
// tracked by ASYNCcnt. Generic LDS pointer's low 32 bits are the LDS address.
__device__ __forceinline__ void async_ld_b128(void* sdst, const float* gsrc) {
  unsigned lds = (unsigned)(unsigned long long)sdst;
  asm volatile("global_load_async_to_lds_b128 %0, %1, off"
               :: "v"(lds), "v"(gsrc) : "memory");
}
#define S_WAIT_ASYNC(N) asm volatile("s_wait_asynccnt %0" :: "i"(N) : "memory")

// ---------------------------------------------------------------- WMMA helpers
__device__ __forceinline__ v8f wmma4(v2f a, v2f b, v8f c) {
  // V_WMMA_F32_16X16X4_F32 : D = A(16x4) x B(4x16) + C(16x16), fp32
  return __builtin_amdgcn_wmma_f32_16x16x4_f32(false, a, false, b, (short)0, c, false, false);
}
__device__ __forceinline__ v8f v8zero() {
  v8f z;
#pragma unroll
  for (int i = 0; i < 8; ++i) z[i] = 0.f;
  return z;
}

__device__ __forceinline__ float mel_freq(int j) {
  // FREQS[j] = (360/11) * (21^(j/31) - 1)
  return (360.0f/11.0f) * (powf(21.0f, (float)j * (1.0f/31.0f)) - 1.0f);
}

__device__ __forceinline__ void rms_row64(const float* __restrict__ in, float* __restrict__ out,
                                          const float* __restrict__ g) {
  float s = 0.f;
#pragma unroll
  for (int d = 0; d < 64; ++d) { float v = in[d]; s += v*v; }
  float r = rsqrtf(s * (1.0f/64.0f) + EPSF);
#pragma unroll
  for (int d = 0; d < 64; ++d) out[d] = in[d] * r * g[d];
}

// C(lds) = A(lds 64x64) @ W(global 64x64) + bias ; wave computes rows [16*wave,16*wave+16)
__device__ __forceinline__ void gemm64_AW(const float* sA, int lda,
                                          const float* __restrict__ W, const float* __restrict__ bias,
                                          float* sC, int ldc, int wave, int lane) {
  const int l = lane & 15, kh = (lane >> 4) << 1, m0 = wave * 16;
  v8f acc[4]; 
#pragma unroll
  for (int nt = 0; nt < 4; ++nt) acc[nt] = v8zero();
#pragma unroll
  for (int k = 0; k < 64; k += 4) {
    v2f a; a.x = sA[(m0+l)*lda + k + kh]; a.y = sA[(m0+l)*lda + k + kh + 1];
#pragma unroll
    for (int nt = 0; nt < 4; ++nt) {
      v2f b; b.x = W[(k+kh)*64 + nt*16 + l]; b.y = W[(k+kh+1)*64 + nt*16 + l];
      acc[nt] = wmma4(a, b, acc[nt]);
    }
  }
  const int rb = m0 + ((lane >= 16) ? 8 : 0);
#pragma unroll
  for (int nt = 0; nt < 4; ++nt) {
    float bv = bias[nt*16 + l];
#pragma unroll
    for (int r = 0; r < 8; ++r) sC[(rb+r)*ldc + nt*16 + l] = acc[nt][r] + bv;
  }
}

// C(lds) = scale * A(lds 64x64) @ B(lds 64x64)^T   (logits: C[q][k']=sum_d A[q][d]*B[k'][d])
__device__ __forceinline__ void gemm64_ABt_scale(const float* sA, int lda, const float* sB, int ldb,
                                                 float scale, float* sC, int ldc, int wave, int lane) {
  const int l = lane & 15, kh = (lane >> 4) << 1, m0 = wave * 16;
  v8f acc[4];
#pragma unroll
  for (int nt = 0; nt < 4; ++nt) acc[nt] = v8zero();
#pragma unroll
  for (int k = 0; k < 64; k += 4) {
    v2f a; a.x = sA[(m0+l)*lda + k + kh]; a.y = sA[(m0+l)*lda + k + kh + 1];
#pragma unroll
    for (int nt = 0; nt < 4; ++nt) {
      v2f b; b.x = sB[(nt*16+l)*ldb + k + kh]; b.y = sB[(nt*16+l)*ldb + k + kh + 1];
      acc[nt] = wmma4(a, b, acc[nt]);
    }
  }
  const int rb = m0 + ((lane >= 16) ? 8 : 0);
#pragma unroll
  for (int nt = 0; nt < 4; ++nt)
#pragma unroll
    for (int r = 0; r < 8; ++r) sC[(rb+r)*ldc + nt*16 + l] = acc[nt][r] * scale;
}

// C(lds) {=, +=} A(lds 64x64) @ B(lds 64x64)
template<bool ACC>
__device__ __forceinline__ void gemm64_AB(const float* sA, int lda, const float* sB, int ldb,
                                          float* sC, int ldc, int wave, int lane) {
  const int l = lane & 15, kh = (lane >> 4) << 1, m0 = wave * 16;
  v8f acc[4];
#pragma unroll
  for (int nt = 0; nt < 4; ++nt) acc[nt] = v8zero();
#pragma unroll
  for (int k = 0; k < 64; k += 4) {
    v2f a; a.x = sA[(m0+l)*lda + k + kh]; a.y = sA[(m0+l)*lda + k + kh + 1];
#pragma unroll
    for (int nt = 0; nt < 4; ++nt) {
      v2f b; b.x = sB[(k+kh)*ldb + nt*16 + l]; b.y = sB[(k+kh+1)*ldb + nt*16 + l];
      acc[nt] = wmma4(a, b, acc[nt]);
    }
  }
  const int rb = m0 + ((lane >= 16) ? 8 : 0);
#pragma unroll
  for (int nt = 0; nt < 4; ++nt)
#pragma unroll
    for (int r = 0; r < 8; ++r) {
      if (ACC) sC[(rb+r)*ldc + nt*16 + l] += acc[nt][r];
      else     sC[(rb+r)*ldc + nt*16 + l]  = acc[nt][r];
    }
}

// ---------------------------------------------------------------- simple kernels
__global__ void zero_kernel(float* p, int n) {
  int i = blockIdx.x*32 + threadIdx.x;
  if (i < n) p[i] = 0.f;
}

// y[row] = x[row] * rsqrt(mean(x^2)+eps) * g   (cols = 1024)
__global__ void rms_kernel(const float* __restrict__ x, const float* __restrict__ g,
                           float* __restrict__ y, int cols) {
  __shared__ float red[256];
  int row = blockIdx.x;
  const float* xr = x + (size_t)row*cols;
  float* yr = y + (size_t)row*cols;
  float s = 0.f;
  for (int i = threadIdx.x; i < cols; i += 256) { float v = xr[i]; s += v*v; }
  red[threadIdx.x] = s; __syncthreads();
  for (int k = 128; k > 0; k >>= 1) { if (threadIdx.x < k) red[threadIdx.x] += red[threadIdx.x+k]; __syncthreads(); }
  float r = rsqrtf(red[0]/(float)cols + EPSF);
  for (int i = threadIdx.x; i < cols; i += 256) yr[i] = xr[i]*r*g[i];
}

// interleaved-pair RoPE in place: position = row % CTX, pairs within each head
__global__ void rope_kernel(float* __restrict__ buf, int ldbc) {
  int idx = blockIdx.x*256 + threadIdx.x;
  if (idx >= ROWS*HEADS*(HD/2)) return;
  int pair = idx & 31;
  int h = (idx >> 5) & 15;
  int row = idx >> 9;
  int c = row % CTX;
  float ang = (float)c * mel_freq(pair);
  float cs = cosf(ang), sn = sinf(ang);
  float* p = buf + (size_t)row*ldbc + h*HD + 2*pair;
  float x0 = p[0], x1 = p[1];
  p[0] = x0*cs - x1*sn;
  p[1] = x0*sn + x1*cs;
}

// curq += attn ; optional diff accumulation |attn - prev|
__global__ void update_kernel(float* __restrict__ curq, const float* __restrict__ attn,
                              const float* __restrict__ prev, float* diffsum, int n) {
  __shared__ float red[256];
  int i = blockIdx.x*256 + threadIdx.x;
  float part = 0.f;
  if (i < n) {
    float a = attn[i];
    curq[i] += a;
    if (prev) part = fabsf(a - prev[i]);
  }
  if (diffsum) {
    red[threadIdx.x] = part; __syncthreads();
    for (int k = 128; k > 0; k >>= 1) { if (threadIdx.x < k) red[threadIdx.x] += red[threadIdx.x+k]; __syncthreads(); }
    if (threadIdx.x == 0) atomicAdd(diffsum, red[0]);
  }
}

// out = (mean_diff < threshold) ? a1 : a2
__global__ void select_kernel(const float* __restrict__ a1, const float* __restrict__ a2,
                              const float* __restrict__ diffsum, const float* __restrict__ thr,
                              float* __restrict__ out, int n) {
  int i = blockIdx.x*256 + threadIdx.x;
  if (i >= n) return;
  bool first = (diffsum[0] * NUMEL_INV) < thr[0];
  out[i] = first ? a1[i] : a2[i];
}

// ---------------------------------------------------------------- big GEMM: C = A@W + bias
// M,N multiples of 64; K multiple of 32. grid=(M/64,N/64), block=128 (4 waves).
// Double-buffered async-to-LDS staging: compute chunk i overlaps DMA of chunk i+1.
#define KC    32
#define SA_LD 36            // 144B rows -> every b128 chunk 16B aligned
#define SW_LD 68            // 272B rows
#define SA_SZ (64*SA_LD)
#define SW_SZ (KC*SW_LD)

__global__ __launch_bounds__(128) void gemm_bias_kernel(
    const float* __restrict__ A, const float* __restrict__ W, const float* __restrict__ bias,
    float* __restrict__ C, int M, int N, int K) {
  __shared__ __align__(16) float sA[2*SA_SZ];
  __shared__ __align__(16) float sW[2*SW_SZ];
  const int t = threadIdx.x, wave = t >> 5, lane = t & 31;
  const int m0 = blockIdx.x*64, n0 = blockIdx.y*64;
  const int l = lane & 15, kh = (lane >> 4) << 1, mw = wave*16;
  v8f acc[4];
#pragma unroll
  for (int nt = 0; nt < 4; ++nt) acc[nt] = v8zero();

  // issue async copies for one K-chunk: A 64xKC (4 b128/thread) + W KCx64 (4 b128/thread)
  auto issue_chunk = [&](int buf, int kc) {
#pragma unroll
    for (int j = 0; j < 4; ++j) {                        // A tile: 512 float4
      int i = t + j*128;
      int r = i >> 3, c4 = (i & 7) << 2;
      async_ld_b128(&sA[buf*SA_SZ + r*SA_LD + c4], A + (size_t)(m0+r)*K + kc + c4);
    }
#pragma unroll
    for (int j = 0; j < 4; ++j) {                        // W tile: 512 float4
      int i = t + j*128;
      int r = i >> 4, c4 = (i & 15) << 2;
      async_ld_b128(&sW[buf*SW_SZ + r*SW_LD + c4], W + (size_t)(kc+r)*N + n0 + c4);
    }
  };

  const int nch = K / KC;
  issue_chunk(0, 0);
  for (int ch = 0; ch < nch; ++ch) {
    const int cur = ch & 1;
    if (ch + 1 < nch) {
      issue_chunk(cur ^ 1, (ch+1)*KC);
      S_WAIT_ASYNC(8);       // async loads retire in order: oldest 8 (chunk ch) are done
    } else {
      S_WAIT_ASYNC(0);
    }
    __syncthreads();
    const float* a_ = sA + cur*SA_SZ;
    const float* w_ = sW + cur*SW_SZ;
#pragma unroll
    for (int kk = 0; kk < KC; kk += 4) {
      v2f a; a.x = a_[(mw+l)*SA_LD + kk + kh]; a.y = a_[(mw+l)*SA_LD + kk + kh + 1];
#pragma unroll
      for (int nt = 0; nt < 4; ++nt) {
        v2f b; b.x = w_[(kk+kh)*SW_LD + nt*16 + l]; b.y = w_[(kk+kh+1)*SW_LD + nt*16 + l];
        acc[nt] = wmma4(a, b, acc[nt]);
      }
    }
    __syncthreads();         // all waves done reading buf `cur` before it is refilled
  }
  const int rb = m0 + mw + ((lane >= 16) ? 8 : 0);
#pragma unroll
  for (int nt = 0; nt < 4; ++nt) {
    int col = n0 + nt*16 + l;
    float bv = bias[col];
#pragma unroll
    for (int r = 0; r < 8; ++r) C[(size_t)(rb+r)*N + col] = acc[nt][r] + bv;
  }
}

// ---------------------------------------------------------------- per-head 64x64 projection
// rows enumerated as (bc,h): addr = bc*ldbc + h*64 + d. M = ROWS*HEADS, grid = M/64
__global__ __launch_bounds__(128) void proj_hd_kernel(
    const float* __restrict__ A, int ldbc_a,
    const float* __restrict__ W, const float* __restrict__ bias,
    float* __restrict__ C, int ldbc_c) {
  __shared__ __align__(16) float sA[64*68];
  __shared__ __align__(16) float sW[64*68];
  const int t = threadIdx.x, wave = t >> 5, lane = t & 31;
  const int r0 = blockIdx.x * 64;

#pragma unroll
  for (int j = 0; j < 8; ++j) {                          // A tile: 1024 float4 async
    int i = t + j*128;
    int rl = i >> 4, c4 = (i & 15) << 2;
    int r = r0 + rl, bc = r >> 4, hh = r & 15;
    async_ld_b128(&sA[rl*68 + c4], A + (size_t)bc*ldbc_a + hh*64 + c4);
  }
#pragma unroll
  for (int j = 0; j < 8; ++j) {                          // W 64x64: 1024 float4 async
    int i = t + j*128;
    int rl = i >> 4, c4 = (i & 15) << 2;
    async_ld_b128(&sW[rl*68 + c4], W + rl*64 + c4);
  }
  S_WAIT_ASYNC(0);
  __syncthreads();

  const int l = lane & 15, kh = (lane >> 4) << 1, mw = wave*16;
  v8f acc[4];
#pragma unroll
  for (int nt = 0; nt < 4; ++nt) acc[nt] = v8zero();
#pragma unroll
  for (int k = 0; k < 64; k += 4) {
    v2f a; a.x = sA[(mw+l)*68 + k + kh]; a.y = sA[(mw+l)*68 + k + kh + 1];
#pragma unroll
    for (int nt = 0; nt < 4; ++nt) {
      v2f b; b.x = sW[(k+kh)*68 + nt*16 + l]; b.y = sW[(k+kh+1)*68 + nt*16 + l];
      acc[nt] = wmma4(a, b, acc[nt]);
    }
  }
  const int rbl = mw + ((lane >= 16) ? 8 : 0);
#pragma unroll
  for (int nt = 0; nt < 4; ++nt) {
    float bv = bias[nt*16 + l];
#pragma unroll
    for (int r = 0; r < 8; ++r) {
      int rg = r0 + rbl + r, bc = rg >> 4, hh = rg & 15;
      C[(size_t)bc*ldbc_c + hh*64 + nt*16 + l] = acc[nt][r] + bv;
    }
  }
}

// ---------------------------------------------------------------- stage-1 windowed focus
// grid=(NWIN, HEADS), block=128. Q: [3072,1024], KV: [3072,2048] (k cols 0..1023, v 1024..2047)
__global__ __launch_bounds__(128) void win_focus_kernel(
    const float* __restrict__ Qbuf, const float* __restrict__ KVbuf,
    const float* __restrict__ wq_hd, const float* __restrict__ bq_hd,
    const float* __restrict__ wk_hd, const float* __restrict__ bk_hd,
    const float* __restrict__ wv_hd, const float* __restrict__ bv_hd,
    const float* __restrict__ g_ln, const float* __restrict__ temp,
    float* __restrict__ out1, float* __restrict__ out2, float* __restrict__ diffsum) {
  extern __shared__ float smem[];
  const int LD = 65, MSZ = 64*65;
  float* sCQ   = smem;
  float* sKa   = sCQ  + MSZ;
  float* sVa   = sKa  + MSZ;
  float* sKb   = sVa  + MSZ;
  float* sVb   = sKb  + MSZ;
  float* sQl   = sVb  + MSZ;
  float* sKn   = sQl  + MSZ;
  float* sS    = sKn  + MSZ;
  float* sO    = sS   + MSZ;
  float* sPrev = sO   + MSZ;
  float* sfreq = sPrev + MSZ;   // 32
  float* ssum  = sfreq + 32;    // 64
  float* sred  = ssum + 64;     // 128

  const int t = threadIdx.x, wave = t >> 5, lane = t & 31;
  const int w = blockIdx.x, h = blockIdx.y;
  const int b = w / WN, wi = w % WN;
  const int qrow0 = b*CTX + wi*64;
  int kst = wi*64 - (HD/HEADS);                 // win*i - win//head
  if (kst < 0) kst = 0;
  const int krow0 = b*CTX + kst;
  const float scale = 0.125f / temp[0];

  if (t < 32) sfreq[t] = mel_freq(t);
  __syncthreads();

  if (t < 64) {                                 // load + local-position RoPE
    const float* qsrc = Qbuf + (size_t)(qrow0 + t)*DIMS + h*HD;
    const float* ksrc = KVbuf + (size_t)(krow0 + t)*(2*DIMS) + h*HD;
    const float* vsrc = ksrc + DIMS;
#pragma unroll
    for (int j = 0; j < 32; ++j) {
      float ang = (float)t * sfreq[j];
      float cs = cosf(ang), sn = sinf(ang);
      float q0 = qsrc[2*j], q1 = qsrc[2*j+1];
      sCQ[t*LD + 2*j]   = q0*cs - q1*sn;
      sCQ[t*LD + 2*j+1] = q0*sn + q1*cs;
      float k0 = ksrc[2*j], k1 = ksrc[2*j+1];
      sKa[t*LD + 2*j]   = k0*cs - k1*sn;
      sKa[t*LD + 2*j+1] = k0*sn + k1*cs;
    }
#pragma unroll
    for (int d = 0; d < 64; ++d) sVa[t*LD + d] = vsrc[d];
  }
  __syncthreads();

  float* pKin = sKa; float* pVin = sVa; float* pKl = sKb; float* pVl = sVb;

  for (int it = 0; it < 3; ++it) {
    gemm64_AW(sCQ,  LD, wq_hd, bq_hd, sQl, LD, wave, lane);
    gemm64_AW(pKin, LD, wk_hd, bk_hd, pKl, LD, wave, lane);
    gemm64_AW(pVin, LD, wv_hd, bv_hd, pVl, LD, wave, lane);
    __syncthreads();
    if (t < 64) {
      rms_row64(sQl + t*LD, sQl + t*LD, g_ln);  // qn (in place, raw ql not reused)
      rms_row64(pKl + t*LD, sKn + t*LD, g_ln);  // kn (keep raw kl for next iter)
    }
    __syncthreads();
    gemm64_ABt_scale(sQl, LD, sKn, LD, scale, sS, LD, wave, lane);
    __syncthreads();
    if (t < 64) {                               // full softmax over 64 keys
      float* row = sS + t*LD;
      float m = -1e30f;
      for (int j = 0; j < 64; ++j) m = fmaxf(m, row[j]);
      float s = 0.f;
      for (int j = 0; j < 64; ++j) { float p = expf(row[j] - m); row[j] = p; s += p; }
      ssum[t] = s;
    }
    __syncthreads();
    gemm64_AB<false>(sS, LD, pVl, LD, sO, LD, wave, lane);
    __syncthreads();
    if (t < 64) {
      float inv = 1.0f / ssum[t];
      for (int d = 0; d < 64; ++d) sO[t*LD + d] *= inv;
    }
    __syncthreads();

    if (it == 0) {
      for (int i = t; i < 4096; i += 128) {
        int r = i >> 6, c = i & 63;
        float v = sO[r*LD + c];
        sPrev[r*LD + c] = v;
        sCQ[r*LD + c] += v;
      }
    } else if (it == 1) {
      float part = 0.f;
      for (int i = t; i < 4096; i += 128) {
        int r = i >> 6, c = i & 63;
        float v = sO[r*LD + c];
        part += fabsf(v - sPrev[r*LD + c]);
        sCQ[r*LD + c] += v;
      }
      sred[t] = part; __syncthreads();
      for (int k = 64; k > 0; k >>= 1) { if (t < k) sred[t] += sred[t+k]; __syncthreads(); }
      if (t == 0) atomicAdd(diffsum, sred[0]);
      if (t < 64) {
        float* dst = out1 + (size_t)(qrow0 + t)*DIMS + h*HD;
        for (int d = 0; d < 64; ++d) dst[d] = sO[t*LD + d];
      }
    } else {
      if (t < 64) {
        float* dst = out2 + (size_t)(qrow0 + t)*DIMS + h*HD;
        for (int d = 0; d < 64; ++d) dst[d] = sO[t*LD + d];
      }
    }
    __syncthreads();
    float* tk = pKin; pKin = pKl; pKl = tk;     // k_cur <- kl
    float* tv = pVin; pVin = pVl; pVl = tv;     // v_cur <- vl
  }
}

// ---------------------------------------------------------------- stage-2 flash attention
// grid=(CTX/64, HEADS, NB), block=128. Buffers in [b,c,h,d] layout (flat 1024/row).
__global__ __launch_bounds__(128) void flash_attn_kernel(
    const float* __restrict__ ql, const float* __restrict__ kl, const float* __restrict__ vl,
    const float* __restrict__ g_ln, const float* __restrict__ temp, float* __restrict__ outb) {
  extern __shared__ float smem[];
  const int LD = 65, MSZ = 64*65;
  float* sQ  = smem;
  float* sK  = sQ + MSZ;
  float* sV  = sK + MSZ;
  float* sS  = sV + MSZ;
  float* sO  = sS + MSZ;
  float* sm  = sO + MSZ;   // 64
  float* sl  = sm + 64;    // 64
  float* sal = sl + 64;    // 64

  const int t = threadIdx.x, wave = t >> 5, lane = t & 31;
  const int qt = blockIdx.x, h = blockIdx.y, b = blockIdx.z;
  const float scale = 0.125f / temp[0];

  if (t < 64) {
    const float* src = ql + (size_t)(b*CTX + qt*64 + t)*DIMS + h*HD;
    rms_row64(src, sQ + t*LD, g_ln);
    sm[t] = -1e30f; sl[t] = 0.f;
  }
  for (int i = t; i < 4096; i += 128) sO[(i>>6)*LD + (i&63)] = 0.f;
  __syncthreads();

  for (int kt = 0; kt < WN; ++kt) {             // 24 key tiles of 64
    if (t < 64) {
      const float* ksrc = kl + (size_t)(b*CTX + kt*64 + t)*DIMS + h*HD;
      rms_row64(ksrc, sK + t*LD, g_ln);
      const float* vsrc = vl + (size_t)(b*CTX + kt*64 + t)*DIMS + h*HD;
      for (int d = 0; d < 64; ++d) sV[t*LD + d] = vsrc[d];
    }
    __syncthreads();
    gemm64_ABt_scale(sQ, LD, sK, LD, scale, sS, LD, wave, lane);
    __syncthreads();
    if (t < 64) {                               // online softmax bookkeeping
      float* row = sS + t*LD;
      float mloc = -1e30f;
      for (int j = 0; j < 64; ++j) mloc = fmaxf(mloc, row[j]);
      float mold = sm[t], mnew = fmaxf(mold, mloc);
      float alpha = expf(mold - mnew);
      float s = 0.f;
      for (int j = 0; j < 64; ++j) { float p = expf(row[j] - mnew); row[j] = p; s += p; }
      sm[t] = mnew; sl[t] = sl[t]*alpha + s; sal[t] = alpha;
    }
    __syncthreads();
    for (int i = t; i < 4096; i += 128) {
      int r = i >> 6, c = i & 63;
      sO[r*LD + c] *= sal[r];
    }
    __syncthreads();
    gemm64_AB<true>(sS, LD, sV, LD, sO, LD, wave, lane);   // O += P@V
    __syncthreads();
  }
  if (t < 64) {
    float inv = 1.0f / sl[t];
    float* dst = outb + (size_t)(b*CTX + qt*64 + t)*DIMS + h*HD;
    for (int d = 0; d < 64; ++d) dst[d] = sO[t*LD + d] * inv;
  }
}

// ---------------------------------------------------------------- host driver
extern "C" void kernel_launch(void* const* d_in, const int* in_sizes, int n_in,
                              void* d_out, int out_size, void* d_ws, size_t ws_size,
                              hipStream_t stream) {
  (void)in_sizes; (void)n_in; (void)out_size; (void)ws_size;
  const float* x     = (const float*)d_in[0];
  const float* xa    = (const float*)d_in[1];
  const float* g_q   = (const float*)d_in[2];
  const float* wq    = (const float*)d_in[3];
  const float* bq    = (const float*)d_in[4];
  const float* g_kv  = (const float*)d_in[5];
  const float* wkv   = (const float*)d_in[6];
  const float* bkv   = (const float*)d_in[7];
  const float* g_ln  = (const float*)d_in[8];
  const float* wq_hd = (const float*)d_in[9];
  const float* bq_hd = (const float*)d_in[10];
  const float* wk_hd = (const float*)d_in[11];
  const float* bk_hd = (const float*)d_in[12];
  const float* wv_hd = (const float*)d_in[13];
  const float* bv_hd = (const float*)d_in[14];
  const float* thr   = (const float*)d_in[15];
  const float* temp  = (const float*)d_in[16];
  const float* wout  = (const float*)d_in[17];
  const float* bout  = (const float*)d_in[18];
  float* out = (float*)d_out;

  const size_t U = (size_t)ROWS * DIMS;         // 3,145,728 floats
  float* ws   = (float*)d_ws;
  float* diff = ws;                             // diff[0]=slide(x), diff[1]=slide(xa), diff[2]=final
  float* R1   = ws + 256;
  float* QX   = R1  + U;
  float* KVX  = QX  + U;        // 2U (stage1 kv of x, reused for stage2 kv of xas)
  float* QA   = KVX + 2*U;
  float* KVA  = QA  + U;        // 2U
  float* O1   = KVA + 2*U;
  float* O2   = O1  + U;
  float* XS   = O2  + U;
  float* XAS  = XS  + U;
  float* QS   = XAS + U;        // stage-2 q, becomes curq (updated in place)
  float* KL0  = QS  + U;
  float* KL1  = KL0 + U;
  float* VL0  = KL1 + U;
  float* VL1  = VL0 + U;
  // aliases onto dead buffers:
  float* A0   = QX;             // attn iter0 (QX dead after win_focus(x))
  float* A1   = O1;             // attn iter1
  float* A2   = O2;             // attn iter2
  float* QL   = QA;             // stage-2 ql scratch (QA dead after win_focus(xa))
  float* FSEL = R1;             // selected final focus output

  const int N = (int)U;                                   // 3,145,728 elements
  const int EB = (N + 255) / 256;                         // elementwise blocks
  const size_t WIN_SMEM   = (size_t)(10*64*65 + 32 + 64 + 128) * sizeof(float);
  const size_t FLASH_SMEM = (size_t)(5*64*65 + 192) * sizeof(float);
  const int ROPE_B = (ROWS*HEADS*(HD/2) + 255) / 256;

  zero_kernel<<<1, 32, 0, stream>>>(diff, 16);

  // ---- stage 1: slide_win(x) ----
  rms_kernel<<<ROWS, 256, 0, stream>>>(x, g_q, R1, DIMS);
  gemm_bias_kernel<<<dim3(ROWS/64, DIMS/64), 128, 0, stream>>>(R1, wq, bq, QX, ROWS, DIMS, DIMS);
  rms_kernel<<<ROWS, 256, 0, stream>>>(x, g_kv, R1, DIMS);
  gemm_bias_kernel<<<dim3(ROWS/64, 2*DIMS/64), 128, 0, stream>>>(R1, wkv, bkv, KVX, ROWS, 2*DIMS, DIMS);
  win_focus_kernel<<<dim3(NWIN, HEADS), 128, WIN_SMEM, stream>>>(
      QX, KVX, wq_hd, bq_hd, wk_hd, bk_hd, wv_hd, bv_hd, g_ln, temp, O1, O2, diff + 0);
  select_kernel<<<EB, 256, 0, stream>>>(O1, O2, diff + 0, thr, XS, N);

  // ---- stage 1: slide_win(xa) ----
  rms_kernel<<<ROWS, 256, 0, stream>>>(xa, g_q, R1, DIMS);
  gemm_bias_kernel<<<dim3(ROWS/64, DIMS/64), 128, 0, stream>>>(R1, wq, bq, QA, ROWS, DIMS, DIMS);
  rms_kernel<<<ROWS, 256, 0, stream>>>(xa, g_kv, R1, DIMS);
  gemm_bias_kernel<<<dim3(ROWS/64, 2*DIMS/64), 128, 0, stream>>>(R1, wkv, bkv, KVA, ROWS, 2*DIMS, DIMS);
  win_focus_kernel<<<dim3(NWIN, HEADS), 128, WIN_SMEM, stream>>>(
      QA, KVA, wq_hd, bq_hd, wk_hd, bk_hd, wv_hd, bv_hd, g_ln, temp, O1, O2, diff + 1);
  select_kernel<<<EB, 256, 0, stream>>>(O1, O2, diff + 1, thr, XAS, N);

  // ---- stage 2: focus(xs, xas), eff = 1536 ----
  rms_kernel<<<ROWS, 256, 0, stream>>>(XS, g_q, R1, DIMS);
  gemm_bias_kernel<<<dim3(ROWS/64, DIMS/64), 128, 0, stream>>>(R1, wq, bq, QS, ROWS, DIMS, DIMS);
  rope_kernel<<<ROPE_B, 256, 0, stream>>>(QS, DIMS);
  rms_kernel<<<ROWS, 256, 0, stream>>>(XAS, g_kv, R1, DIMS);
  gemm_bias_kernel<<<dim3(ROWS/64, 2*DIMS/64), 128, 0, stream>>>(R1, wkv, bkv, KVX, ROWS, 2*DIMS, DIMS);
  rope_kernel<<<ROPE_B, 256, 0, stream>>>(KVX, 2*DIMS);   // k half only (cols 0..1023)

  const int PB = ROWS*HEADS/64;                           // 768 proj blocks
  // iter 0
  proj_hd_kernel<<<PB, 128, 0, stream>>>(QS, DIMS, wq_hd, bq_hd, QL, DIMS);
  proj_hd_kernel<<<PB, 128, 0, stream>>>(KVX, 2*DIMS, wk_hd, bk_hd, KL0, DIMS);
  proj_hd_kernel<<<PB, 128, 0, stream>>>(KVX + DIMS, 2*DIMS, wv_hd, bv_hd, VL0, DIMS);
  flash_attn_kernel<<<dim3(CTX/64, HEADS, NB), 128, FLASH_SMEM, stream>>>(QL, KL0, VL0, g_ln, temp, A0);
  update_kernel<<<EB, 256, 0, stream>>>(QS, A0, nullptr, nullptr, N);
  // iter 1
  proj_hd_kernel<<<PB, 128, 0, stream>>>(QS, DIMS, wq_hd, bq_hd, QL, DIMS);
  proj_hd_kernel<<<PB, 128, 0, stream>>>(KL0, DIMS, wk_hd, bk_hd, KL1, DIMS);
  proj_hd_kernel<<<PB, 128, 0, stream>>>(VL0, DIMS, wv_hd, bv_hd, VL1, DIMS);
  flash_attn_kernel<<<dim3(CTX/64, HEADS, NB), 128, FLASH_SMEM, stream>>>(QL, KL1, VL1, g_ln, temp, A1);
  update_kernel<<<EB, 256, 0, stream>>>(QS, A1, A0, diff + 2, N);
  // iter 2
  proj_hd_kernel<<<PB, 128, 0, stream>>>(QS, DIMS, wq_hd, bq_hd, QL, DIMS);
  proj_hd_kernel<<<PB, 128, 0, stream>>>(KL1, DIMS, wk_hd, bk_hd, KL0, DIMS);
  proj_hd_kernel<<<PB, 128, 0, stream>>>(VL1, DIMS, wv_hd, bv_hd, VL0, DIMS);
  flash_attn_kernel<<<dim3(CTX/64, HEADS, NB), 128, FLASH_SMEM, stream>>>(QL, KL0, VL0, g_ln, temp, A2);
  select_kernel<<<EB, 256, 0, stream>>>(A1, A2, diff + 2, thr, FSEL, N);

  // ---- output projection ----
  gemm_bias_kernel<<<dim3(ROWS/64, DIMS/64), 128, 0, stream>>>(FSEL, wout, bout, out, ROWS, DIMS, DIMS);
}